// GAT_2L_60868276519457
// MI455X (gfx1250) — compile-verified
//
#include <hip/hip_runtime.h>
#include <hip/hip_bf16.h>

typedef __attribute__((ext_vector_type(2))) float v2f;
typedef __attribute__((ext_vector_type(8))) float v8f;

#define NEG_SLOPE 0.2f

// ---------------------------------------------------------------------------
// GEMM: C[M,128] = A[M,128] * B[128,128], f32, via V_WMMA_F32_16X16X4_F32.
// One wave -> one 16x16 tile. Block = 256 thr = 8 waves = all 8 N-tiles.
// ---------------------------------------------------------------------------
__global__ __launch_bounds__(256)
void gat_gemm_wmma_f32(const float* __restrict__ A, const float* __restrict__ B,
                       float* __restrict__ C, int M) {
    const int lane = threadIdx.x & 31;
    const int wave = threadIdx.x >> 5;       // 0..7 -> n-tile
    const int m0   = blockIdx.x * 16;
    const int n0   = wave * 16;
    const int half = lane >> 4;              // 0: lanes 0-15, 1: lanes 16-31
    const int l15  = lane & 15;

    int row = m0 + l15;
    if (row >= M) row = M - 1;               // clamp; padded rows never stored

    // A fragment: lane half 0 holds K={k,k+1}, half 1 holds K={k+2,k+3} of row.
    const float* arow = A + (size_t)row * 128 + 2 * half;
    // B fragment: col = n0+l15; half 0 -> K rows {k,k+1}, half 1 -> {k+2,k+3}.
    const float* bcol = B + (size_t)(2 * half) * 128 + n0 + l15;

    v8f acc = {};
#pragma unroll
    for (int k = 0; k < 128; k += 4) {
        v2f a;
        a.x = arow[k];
        a.y = arow[k + 1];
        v2f b;
        b.x = bcol[(size_t)k * 128];
        b.y = bcol[(size_t)(k + 1) * 128];
        acc = __builtin_amdgcn_wmma_f32_16x16x4_f32(
            /*neg_a=*/false, a, /*neg_b=*/false, b,
            /*c_mod=*/(short)0, acc, /*reuse_a=*/false, /*reuse_b=*/false);
    }

    // D layout: VGPR r -> row m0 + 8*half + r, col n0 + l15
    const int orow = m0 + half * 8;
    float* cp = C + (size_t)orow * 128 + n0 + l15;
#pragma unroll
    for (int r = 0; r < 8; ++r) {
        if (orow + r < M) cp[(size_t)r * 128] = acc[r];
    }
}

// ---------------------------------------------------------------------------
// Per-node attention dot products: a_src[n] = h[n,:]*att_src, same for dst.
// One wave per node, float4 per lane, shfl_xor reduction (wave32).
// ---------------------------------------------------------------------------
__global__ void gat_attn_dots(const float* __restrict__ h,
                              const float* __restrict__ att_s,
                              const float* __restrict__ att_d,
                              float* __restrict__ a_src,
                              float* __restrict__ a_dst, int N) {
    const int lane = threadIdx.x & 31;
    const int n = blockIdx.x * (blockDim.x >> 5) + (threadIdx.x >> 5);
    if (n >= N) return;
    float4 v = ((const float4*)(h + (size_t)n * 128))[lane];
    float4 s = ((const float4*)att_s)[lane];
    float4 d = ((const float4*)att_d)[lane];
    float ps = v.x * s.x + v.y * s.y + v.z * s.z + v.w * s.w;
    float pd = v.x * d.x + v.y * d.y + v.z * d.z + v.w * d.w;
#pragma unroll
    for (int m = 16; m >= 1; m >>= 1) {
        ps += __shfl_xor(ps, m, 32);
        pd += __shfl_xor(pd, m, 32);
    }
    if (lane == 0) {
        a_src[n] = ps;
        a_dst[n] = pd;
    }
}

__global__ void gat_fill_zero(float* __restrict__ p, size_t n) {
    size_t i = (size_t)blockIdx.x * blockDim.x + threadIdx.x;
    if (i < n) p[i] = 0.0f;
}

__global__ void gat_init_nodes(unsigned int* __restrict__ mkey,
                               float* __restrict__ denom, int N) {
    int i = blockIdx.x * blockDim.x + threadIdx.x;
    if (i < N) {
        mkey[i] = 0u;   // below encode() of any real value (every dst has a self loop)
        denom[i] = 0.0f;
    }
}

__device__ __forceinline__ unsigned int enc_f32(float f) {
    unsigned int u = __float_as_uint(f);
    return (u & 0x80000000u) ? ~u : (u | 0x80000000u);   // order-preserving map
}
__device__ __forceinline__ float dec_f32(unsigned int k) {
    return (k & 0x80000000u) ? __uint_as_float(k & 0x7FFFFFFFu)
                             : __uint_as_float(~k);
}

// Pass 1 over edges: e = leakyrelu(a_src[s] + a_dst[d]); segment max via
// order-preserving uint atomicMax. Self loops are edges [E, E+N).
__global__ void gat_edge_lrelu_max(const long long* __restrict__ ei, long long E,
                                   int N, const float* __restrict__ a_src,
                                   const float* __restrict__ a_dst,
                                   float* __restrict__ ebuf,
                                   unsigned int* __restrict__ mkey) {
    long long t = (long long)blockIdx.x * blockDim.x + threadIdx.x;
    if (t >= E + N) return;
    int s, d;
    if (t < E) { s = (int)ei[t]; d = (int)ei[E + t]; }
    else       { s = d = (int)(t - E); }
    float e = a_src[s] + a_dst[d];
    e = (e > 0.0f) ? e : NEG_SLOPE * e;
    ebuf[t] = e;
    atomicMax(mkey + d, enc_f32(e));
}

__global__ void gat_decode_max(const unsigned int* __restrict__ mkey,
                               float* __restrict__ mf, int N) {
    int i = blockIdx.x * blockDim.x + threadIdx.x;
    if (i < N) mf[i] = dec_f32(mkey[i]);
}

// Pass 2 over edges: ex = exp(e - max[d]); denom[d] += ex (ex overwrites ebuf).
__global__ void gat_edge_exp_sum(const long long* __restrict__ ei, long long E,
                                 int N, float* __restrict__ ebuf,
                                 const float* __restrict__ mf,
                                 float* __restrict__ denom) {
    long long t = (long long)blockIdx.x * blockDim.x + threadIdx.x;
    if (t >= E + N) return;
    int d = (t < E) ? (int)ei[E + t] : (int)(t - E);
    float ex = __expf(ebuf[t] - mf[d]);
    ebuf[t] = ex;
    atomicAdd(denom + d, ex);
}

// Pass 3 over edges: out[d,:] += (ex/denom[d]) * h[s,:].
// One wave per edge; lane handles 4 contiguous floats (32*4 = 128 = full row).
__global__ __launch_bounds__(256)
void gat_edge_aggregate(const long long* __restrict__ ei, long long E, int N,
                        const float* __restrict__ ex,
                        const float* __restrict__ denom,
                        const float* __restrict__ h, float* __restrict__ out) {
    const int lane = threadIdx.x & 31;
    long long t = (long long)blockIdx.x * (blockDim.x >> 5) + (threadIdx.x >> 5);
    if (t >= E + N) return;
    int s, d;
    if (t < E) { s = (int)ei[t]; d = (int)ei[E + t]; }
    else       { s = d = (int)(t - E); }
    float alpha = ex[t] / denom[d];
    float4 v = ((const float4*)(h + (size_t)s * 128))[lane];
    float* o = out + (size_t)d * 128 + lane * 4;
    atomicAdd(o + 0, alpha * v.x);
    atomicAdd(o + 1, alpha * v.y);
    atomicAdd(o + 2, alpha * v.z);
    atomicAdd(o + 3, alpha * v.w);
}

__global__ void gat_bias_relu(const float* __restrict__ acc,
                              const float* __restrict__ bias,
                              float* __restrict__ outp, size_t total) {
    size_t i = (size_t)blockIdx.x * blockDim.x + threadIdx.x;
    if (i < total) {
        float v = acc[i] + bias[i & 127u];
        outp[i] = v > 0.0f ? v : 0.0f;
    }
}

// ---------------------------------------------------------------------------
extern "C" void kernel_launch(void* const* d_in, const int* in_sizes, int n_in,
                              void* d_out, int out_size, void* d_ws, size_t ws_size,
                              hipStream_t stream) {
    const float*     x   = (const float*)d_in[0];
    const long long* ei  = (const long long*)d_in[1];   // int64 [2,E]
    const float*     W1  = (const float*)d_in[2];
    const float*     as1 = (const float*)d_in[3];
    const float*     ad1 = (const float*)d_in[4];
    const float*     b1  = (const float*)d_in[5];
    const float*     W2  = (const float*)d_in[6];
    const float*     as2 = (const float*)d_in[7];
    const float*     ad2 = (const float*)d_in[8];
    const float*     b2  = (const float*)d_in[9];

    const int       N  = in_sizes[0] / 128;
    const long long E  = (long long)(in_sizes[1] / 2);
    const long long ET = E + N;
    const size_t    NF = (size_t)N * 128;

    // Workspace carve-out
    float* ws    = (float*)d_ws;
    float* h     = ws;                    ws += NF;
    float* acc   = ws;                    ws += NF;
    float* hin2  = ws;                    ws += NF;
    float* a_src = ws;                    ws += N;
    float* a_dst = ws;                    ws += N;
    unsigned int* mkey = (unsigned int*)ws; ws += N;
    float* mf    = ws;                    ws += N;
    float* denom = ws;                    ws += N;
    float* ebuf  = ws;                    ws += ET;

    const int edgeBlocks  = (int)((ET + 255) / 256);
    const int edgeWaveBlk = (int)((ET + 7) / 8);        // 8 waves/block
    const int nfBlocks    = (int)((NF + 255) / 256);

    auto layer = [&](const float* xin, const float* W, const float* att_s,
                     const float* att_d, const float* bias, float* relu_out) {
        gat_gemm_wmma_f32<<<(N + 15) / 16, 256, 0, stream>>>(xin, W, h, N);
        gat_attn_dots<<<(N + 3) / 4, 128, 0, stream>>>(h, att_s, att_d,
                                                       a_src, a_dst, N);
        gat_fill_zero<<<nfBlocks, 256, 0, stream>>>(acc, NF);
        gat_init_nodes<<<(N + 255) / 256, 256, 0, stream>>>(mkey, denom, N);
        gat_edge_lrelu_max<<<edgeBlocks, 256, 0, stream>>>(ei, E, N, a_src,
                                                           a_dst, ebuf, mkey);
        gat_decode_max<<<(N + 255) / 256, 256, 0, stream>>>(mkey, mf, N);
        gat_edge_exp_sum<<<edgeBlocks, 256, 0, stream>>>(ei, E, N, ebuf, mf, denom);
        gat_edge_aggregate<<<edgeWaveBlk, 256, 0, stream>>>(ei, E, N, ebuf,
                                                            denom, h, acc);
        gat_bias_relu<<<nfBlocks, 256, 0, stream>>>(acc, bias, relu_out, NF);
    };

    layer(x,    W1, as1, ad1, b1, hin2);
    layer(hin2, W2, as2, ad2, b2, (float*)d_out);
}